// MSGIFSR_72129680769633
// MI455X (gfx1250) — compile-verified
//
#include <hip/hip_runtime.h>
#include <hip/hip_bf16.h>

// ---------------------------------------------------------------------------
// MSGIFSR dual-GAT block for MI455X (gfx1250, wave32, WMMA f32 16x16x4,
// async global->LDS staging, K-pair-interleaved LDS tiles).
// ---------------------------------------------------------------------------

typedef float v2f __attribute__((ext_vector_type(2)));
typedef float v8f __attribute__((ext_vector_type(8)));

static __device__ __forceinline__ v8f wmma_f32(v2f a, v2f b, v8f c) {
  // D = A(16x4) * B(4x16) + C(16x16), full fp32
  return __builtin_amdgcn_wmma_f32_16x16x4_f32(false, a, false, b, (short)0, c,
                                               false, false);
}

// CDNA5 async copy: copies 4 bytes from global gptr to the LDS address in lds_off, tracked by ASYNCcnt.
static __device__ __forceinline__ void async_copy_b32(unsigned lds_off,
                                                      const void* gptr) {
  asm volatile("global_load_async_to_lds_b32 %0, %1, off" ::"v"(lds_off),
               "v"((unsigned long long)(uintptr_t)gptr)
               : "memory");
}
static __device__ __forceinline__ void wait_async0() {
  asm volatile("s_wait_asynccnt 0x0" ::: "memory");
}
// Generic->LDS offset: flat-aperture low 32 bits are the wave LDS byte offset.
static __device__ __forceinline__ unsigned lds_off(const void* p) {
  return (unsigned)(uintptr_t)p;
}

static __device__ __forceinline__ float lrelu(float v) {
  return v >= 0.f ? v : 0.2f * v;
}

// Problem constants
#define NB 64
#define NORD 2
#define NCV 2
#define NS 256
#define ND 256
#define NH 8
#define NHD 32

// Scratch layout (float element offsets inside d_ws)
#define OFF_Y 0ull                                  // [b,i,c,s,d]  16,777,216 f
#define OFF_AS (OFF_Y + 16777216ull)                // [b,i,c,s,h]     524,288 f
#define OFF_AD (OFF_AS + 524288ull)                 // [b,i,c,t,h]     524,288 f
#define OFF_M (OFF_AD + 524288ull)                  // [b,i,c,t,h]     524,288 f
#define OFF_IL (OFF_M + 524288ull)                  // [b,i,c,t,h]     524,288 f
#define OFF_PRE (OFF_IL + 524288ull)                // [b,i,t,d]     8,388,608 f
#define OFF_EB (OFF_PRE + 8388608ull)               // u32 [b,i,c,t,s/32]

// ---------------------------------------------------------------------------
// Kernel 1: y[b,i,c] = x[b,i] @ W[i,c] + bias[i,c]   (WMMA f32 GEMM)
// Block: 256 threads = 8 waves, tile 128(M) x 64(N), K staged 32 at a time.
// Wtile is K-pair interleaved: Wtile[(k/2)*128 + n*2 + (k&1)] = W[k][n] so a
// B fragment (rows k,k+1, col n) is one aligned ds_load_b64.
// ---------------------------------------------------------------------------
__global__ void __launch_bounds__(256) proj_kernel(
    const float* __restrict__ x, const float* __restrict__ Wp,
    const float* __restrict__ bp, float* __restrict__ y) {
  __shared__ float Wtile[32 * 64];
  const int tid = threadIdx.x;
  const int lane = tid & 31, w = tid >> 5;
  const int l15 = lane & 15;
  const bool hi = lane >= 16;

  const int bz = blockIdx.y;  // b*4 + i*2 + c
  const int b = bz >> 2, i = (bz >> 1) & 1, c = bz & 1;
  const int Mtile = blockIdx.x >> 2, Ntile = blockIdx.x & 3;
  const int Mbase = Mtile * 128 + w * 16;
  const int Nbase = Ntile * 64;

  const float* xp = x + (size_t)(b * NORD + i) * NS * ND;
  const float* wp = Wp + (size_t)(i * NCV + c) * ND * ND;
  const int mrow = Mbase + l15;
  const unsigned wt_base = lds_off(Wtile);

  v8f acc[4];
  for (int n = 0; n < 4; ++n)
    for (int r = 0; r < 8; ++r) acc[n][r] = 0.f;

  for (int kc = 0; kc < ND; kc += 32) {
    __syncthreads();  // readers of previous panel done
    for (int j = 0; j < 8; ++j) {
      const int idx = tid + j * 256;           // 0..2047
      const int kr = idx >> 6, nc = idx & 63;  // k-row, col within panel
      const unsigned dst = (unsigned)((kr >> 1) * 128 + nc * 2 + (kr & 1));
      async_copy_b32(wt_base + dst * 4u,
                     wp + (size_t)(kc + kr) * ND + Nbase + nc);
    }
    if (kc + 32 < ND)  // speculative prefetch of the next panel
      __builtin_prefetch(wp + (size_t)(kc + 32) * ND + Nbase + (tid & 31) * 2,
                         0, 1);
    wait_async0();
    __syncthreads();

    for (int k4 = 0; k4 < 8; ++k4) {
      const int kk = kc + k4 * 4 + (hi ? 2 : 0);  // global K index (even)
      const int p = k4 * 2 + (hi ? 1 : 0);        // K-pair within panel
      const float* ap = xp + (size_t)mrow * ND + kk;
      v2f a;
      a.x = ap[0];
      a.y = ap[1];
      for (int n = 0; n < 4; ++n) {
        const float* bpp = &Wtile[p * 128 + (n * 16 + l15) * 2];
        v2f bf;
        bf.x = bpp[0];
        bf.y = bpp[1];
        acc[n] = wmma_f32(a, bf, acc[n]);
      }
    }
  }

  float* yp = y + (size_t)bz * NS * ND;
  const float* bb = bp + (size_t)(i * NCV + c) * ND;
  for (int n = 0; n < 4; ++n) {
    const int col = Nbase + n * 16 + l15;
    const float bias = bb[col];
    for (int r = 0; r < 8; ++r) {
      const int m = Mbase + r + (hi ? 8 : 0);
      yp[(size_t)m * ND + col] = acc[n][r] + bias;
    }
  }
}

// ---------------------------------------------------------------------------
// Kernel 2: per-head source/dest scores a_s, a_d  (one thread / (b,i,c,s,h))
// ---------------------------------------------------------------------------
__global__ void __launch_bounds__(256) score_kernel(
    const float* __restrict__ y, const float* __restrict__ att_src,
    const float* __restrict__ att_dst, float* __restrict__ as_out,
    float* __restrict__ ad_out) {
  const int g = blockIdx.x * 256 + threadIdx.x;  // [b,i,c,s,h], h fastest
  const int h = g & 7;
  const int s = (g >> 3) & 255;
  const int bic = g >> 11;
  const int ic = bic & 3;
  const float* yr = y + ((size_t)bic * NS + s) * ND + h * NHD;
  const float* sv = att_src + ((size_t)ic * NH + h) * NHD;
  const float* dv = att_dst + ((size_t)ic * NH + h) * NHD;
  float sa = 0.f, sd = 0.f;
  for (int d = 0; d < NHD; ++d) {
    const float v = yr[d];
    sa += v * sv[d];
    sd += v * dv[d];
  }
  as_out[g] = sa;
  ad_out[g] = sd;
}

// ---------------------------------------------------------------------------
// Kernel 3: softmax stats m[t,h], 1/l[t,h] over source axis + packed edge bits
// ---------------------------------------------------------------------------
__global__ void __launch_bounds__(256) stats_kernel(
    const int* __restrict__ A, const float* __restrict__ as_in,
    const float* __restrict__ ad_in, float* __restrict__ m_out,
    float* __restrict__ il_out, unsigned* __restrict__ ebits) {
  const int g = blockIdx.x * 256 + threadIdx.x;  // [b,i,c,t,h]
  const int h = g & 7;
  const int t = (g >> 3) & 255;
  const int bic = g >> 11;
  const int c = bic & 1;
  const int bi = bic >> 1;

  const int* Ap = A + (size_t)bi * NS * NS + t;         // + s*NS
  const float* asp = as_in + (size_t)bic * NS * 8 + h;  // + s*8
  const float ad = ad_in[((size_t)bic * NS + t) * 8 + h];

  unsigned bits[8];
  for (int j = 0; j < 8; ++j) bits[j] = 0u;

  float m = -1e30f;
  for (int s = 0; s < NS; ++s) {
    const int v = Ap[(size_t)s * NS];
    const bool e = (c == 0) ? (v == 2 || v == 4) : (v == 3 || v == 4);
    if (e) bits[s >> 5] |= 1u << (s & 31);
    const float q = asp[(size_t)s * 8] + ad;
    const float gq = e ? lrelu(q) : 0.f;
    m = fmaxf(m, gq);
  }
  float sum = 0.f;
  for (int s = 0; s < NS; ++s) {
    const bool e = (bits[s >> 5] >> (s & 31)) & 1u;
    const float q = asp[(size_t)s * 8] + ad;
    const float gq = e ? lrelu(q) : 0.f;
    sum += __expf(gq - m);
  }
  m_out[g] = m;
  il_out[g] = 1.f / sum;
  if (h == 0) {
    unsigned* ep = ebits + ((size_t)bic * NS + t) * 8;
    for (int j = 0; j < 8; ++j) ep[j] = bits[j];
  }
}

// ---------------------------------------------------------------------------
// Kernel 4: fused softmax * V  (WMMA f32): for (b,i,h) accumulate over c:
//   pre[t, h*32+n] += (1/l[t]) * sum_s exp(g(s,t)-m[t]) * Y[s,n] + Y[t,n] + gb
// A-fragments (exp weights) generated in registers from packed edge bits.
// Yt is K-pair interleaved (one ds_load_b64 per B fragment); a_s row staged
// in LDS so the K-loop score reads are a single ds_load_b64 too.
// Block = 256 threads = 8 waves; wave w owns dst rows [w*32, w*32+32).
// ---------------------------------------------------------------------------
__global__ void __launch_bounds__(256) attn_kernel(
    const float* __restrict__ y, const float* __restrict__ as_in,
    const float* __restrict__ ad_in, const float* __restrict__ m_in,
    const float* __restrict__ il_in, const unsigned* __restrict__ ebits,
    const float* __restrict__ gbias, float* __restrict__ preb) {
  __shared__ float Yt[NS * NHD];  // 32 KB head panel, pair-interleaved
  __shared__ float asRow[NS];     // 1 KB per-head source scores
  const int tid = threadIdx.x;
  const int lane = tid & 31, w = tid >> 5;
  const int l15 = lane & 15;
  const bool hi = lane >= 16;

  const int h = blockIdx.x & 7;
  const int i = (blockIdx.x >> 3) & 1;
  const int b = blockIdx.x >> 4;

  const int tb0 = w * 32;
  const int t0 = tb0 + l15;  // A-frag row for M-subtile 0
  const int t1 = t0 + 16;    // A-frag row for M-subtile 1
  const unsigned yt_base = lds_off(Yt);
  const unsigned ar_base = lds_off(asRow);

  v8f pre[2][2];
  for (int ms = 0; ms < 2; ++ms)
    for (int ns = 0; ns < 2; ++ns)
      for (int r = 0; r < 8; ++r) pre[ms][ns][r] = 0.f;

  for (int c = 0; c < NCV; ++c) {
    const int bic = (b * NORD + i) * NCV + c;
    const float* yb = y + (size_t)bic * NS * ND + h * NHD;

    __syncthreads();  // readers of previous panel done
    for (int j = 0; j < 32; ++j) {
      const int idx = tid + j * 256;  // 0..8191
      const int s = idx >> 5, n = idx & 31;
      const unsigned dst = (unsigned)(((s >> 1) * NHD + n) * 2 + (s & 1));
      async_copy_b32(yt_base + dst * 4u, yb + (size_t)s * ND + n);
    }
    // a_s[s=tid, h] for this head
    async_copy_b32(ar_base + (unsigned)tid * 4u,
                   as_in + ((size_t)bic * NS + tid) * 8 + h);
    wait_async0();
    __syncthreads();

    const float ad0 = ad_in[((size_t)bic * NS + t0) * 8 + h];
    const float ad1 = ad_in[((size_t)bic * NS + t1) * 8 + h];
    const float m0 = m_in[((size_t)bic * NS + t0) * 8 + h];
    const float m1 = m_in[((size_t)bic * NS + t1) * 8 + h];
    const unsigned* eb0 = ebits + ((size_t)bic * NS + t0) * 8;
    const unsigned* eb1 = ebits + ((size_t)bic * NS + t1) * 8;

    v8f acc[2][2];
    for (int ms = 0; ms < 2; ++ms)
      for (int ns = 0; ns < 2; ++ns)
        for (int r = 0; r < 8; ++r) acc[ms][ns][r] = 0.f;

    unsigned w0 = 0u, w1 = 0u;
    for (int k = 0; k < NS; k += 4) {
      if ((k & 31) == 0) {
        w0 = eb0[k >> 5];
        w1 = eb1[k >> 5];
      }
      const int kk = k + (hi ? 2 : 0);  // even
      const float* ar = &asRow[kk];     // contiguous pair -> ds_load_b64
      const float as0 = ar[0];
      const float as1 = ar[1];

      // exp-weight A fragments (2 elems / lane / M-subtile)
      const float qa0 = as0 + ad0, qa1 = as1 + ad0;
      const float g00 = ((w0 >> (kk & 31)) & 1u) ? lrelu(qa0) : 0.f;
      const float g01 = ((w0 >> ((kk + 1) & 31)) & 1u) ? lrelu(qa1) : 0.f;
      v2f a0;
      a0.x = __expf(g00 - m0);
      a0.y = __expf(g01 - m0);
      const float qb0 = as0 + ad1, qb1 = as1 + ad1;
      const float g10 = ((w1 >> (kk & 31)) & 1u) ? lrelu(qb0) : 0.f;
      const float g11 = ((w1 >> ((kk + 1) & 31)) & 1u) ? lrelu(qb1) : 0.f;
      v2f a1;
      a1.x = __expf(g10 - m1);
      a1.y = __expf(g11 - m1);

      // B fragments from LDS head panel (pair-interleaved -> ds_load_b64)
      const int p = (k >> 1) + (hi ? 1 : 0);  // == kk/2
      const float* b0p = &Yt[(p * NHD + l15) * 2];
      const float* b1p = &Yt[(p * NHD + 16 + l15) * 2];
      v2f bf0, bf1;
      bf0.x = b0p[0];
      bf0.y = b0p[1];
      bf1.x = b1p[0];
      bf1.y = b1p[1];

      acc[0][0] = wmma_f32(a0, bf0, acc[0][0]);
      acc[0][1] = wmma_f32(a0, bf1, acc[0][1]);
      acc[1][0] = wmma_f32(a1, bf0, acc[1][0]);
      acc[1][1] = wmma_f32(a1, bf1, acc[1][1]);
    }

    // epilogue: row scale 1/l, + skip (projected x) + graph bias
    const float* gb = gbias + (size_t)(i * NCV + c) * ND + h * NHD;
    const float gb0 = gb[l15], gb1 = gb[16 + l15];
    for (int ms = 0; ms < 2; ++ms) {
      const int tb = tb0 + ms * 16;
      for (int r = 0; r < 8; ++r) {
        const int t = tb + r + (hi ? 8 : 0);  // C/D layout row
        const float il = il_in[((size_t)bic * NS + t) * 8 + h];
        const float ysk0 = Yt[((t >> 1) * NHD + l15) * 2 + (t & 1)];
        const float ysk1 = Yt[((t >> 1) * NHD + 16 + l15) * 2 + (t & 1)];
        pre[ms][0][r] += il * acc[ms][0][r] + ysk0 + gb0;
        pre[ms][1][r] += il * acc[ms][1][r] + ysk1 + gb1;
      }
    }
  }

  float* pp = preb + (size_t)(b * NORD + i) * NS * ND + h * NHD;
  for (int ms = 0; ms < 2; ++ms) {
    const int tb = tb0 + ms * 16;
    for (int r = 0; r < 8; ++r) {
      const int t = tb + r + (hi ? 8 : 0);
      pp[(size_t)t * ND + l15] = pre[ms][0][r];
      pp[(size_t)t * ND + 16 + l15] = pre[ms][1][r];
    }
  }
}

// ---------------------------------------------------------------------------
// Kernel 5: cross-order mean + PReLU
// ---------------------------------------------------------------------------
__global__ void __launch_bounds__(256) final_kernel(
    const float* __restrict__ preb, const float* __restrict__ prelu_a,
    float* __restrict__ out) {
  const int g = blockIdx.x * 256 + threadIdx.x;  // over B*S*D
  const int d = g & 255;
  const int t = (g >> 8) & 255;
  const int b = g >> 16;
  const size_t i0 = ((size_t)(b * NORD + 0) * NS + t) * ND + d;
  const size_t i1 = ((size_t)(b * NORD + 1) * NS + t) * ND + d;
  const float p0 = preb[i0], p1 = preb[i1];
  const float mean = 0.5f * (p0 + p1);
  const float a = prelu_a[d];
  const float o0 = p0 + mean, o1 = p1 + mean;
  out[i0] = o0 >= 0.f ? o0 : a * o0;
  out[i1] = o1 >= 0.f ? o1 : a * o1;
}

// ---------------------------------------------------------------------------
extern "C" void kernel_launch(void* const* d_in, const int* in_sizes, int n_in,
                              void* d_out, int out_size, void* d_ws,
                              size_t ws_size, hipStream_t stream) {
  (void)in_sizes;
  (void)n_in;
  (void)out_size;
  (void)ws_size;

  const float* x = (const float*)d_in[0];
  const int* A = (const int*)d_in[1];
  const float* Wp = (const float*)d_in[2];
  const float* bp = (const float*)d_in[3];
  const float* att_src = (const float*)d_in[4];
  const float* att_dst = (const float*)d_in[5];
  const float* gbias = (const float*)d_in[6];
  const float* prelu_a = (const float*)d_in[7];
  float* out = (float*)d_out;

  float* wsf = (float*)d_ws;
  float* yb = wsf + OFF_Y;
  float* as_b = wsf + OFF_AS;
  float* ad_b = wsf + OFF_AD;
  float* m_b = wsf + OFF_M;
  float* il_b = wsf + OFF_IL;
  float* pre_b = wsf + OFF_PRE;
  unsigned* eb_b = (unsigned*)(wsf + OFF_EB);

  // 1) projections: 8 tiles x (B*ORDER*NC = 256) problems
  proj_kernel<<<dim3(8, NB * NORD * NCV), 256, 0, stream>>>(x, Wp, bp, yb);
  // 2) per-head scores: B*ORDER*NC*S*H = 524288 threads
  score_kernel<<<2048, 256, 0, stream>>>(yb, att_src, att_dst, as_b, ad_b);
  // 3) softmax stats + edge bitmask
  stats_kernel<<<2048, 256, 0, stream>>>(A, as_b, ad_b, m_b, il_b, eb_b);
  // 4) fused softmax*V attention GEMM: B*ORDER*H = 1024 blocks
  attn_kernel<<<NB * NORD * NH, 256, 0, stream>>>(yb, as_b, ad_b, m_b, il_b,
                                                  eb_b, gbias, pre_b);
  // 5) order-mean + PReLU: B*S*D / 256 blocks
  final_kernel<<<(NB * NS * ND) / 256, 256, 0, stream>>>(pre_b, prelu_a, out);
}